// GraphSAGE_64175401337058
// MI455X (gfx1250) — compile-verified
//
#include <hip/hip_runtime.h>
#include <hip/hip_bf16.h>

#define N_NODES 50000
#define N_EDGES 800000
#define IN_DIM  64
#define HID_DIM 64
#define OUT_DIM 32

typedef __attribute__((ext_vector_type(2))) float v2f;
typedef __attribute__((ext_vector_type(8))) float v8f;

// ---------------------------------------------------------------------------
// Edge scatter: msg[dst, f] += X[src, f]  (hardware f32 atomics, L2-resident)
// One work-item handles 4 consecutive features of one edge (16 chunks/edge).
// Chunk 0 additionally bumps the per-destination edge count (layer 1 only).
// ---------------------------------------------------------------------------
__global__ __launch_bounds__(256) void sage_scatter(
    const float* __restrict__ X,      // [N_NODES, 64]
    const int*   __restrict__ edges,  // [2, N_EDGES] (src row, dst row)
    float* __restrict__ msg,          // [N_NODES, 64] pre-zeroed
    float* __restrict__ cnt,          // [N_NODES] or nullptr
    int nEdges)
{
    const long long total = (long long)nEdges * 16;
    long long idx = (long long)blockIdx.x * blockDim.x + threadIdx.x;
    const long long stride = (long long)gridDim.x * blockDim.x;
    for (; idx < total; idx += stride) {
        const int e  = (int)(idx >> 4);
        const int c4 = (int)(idx & 15) * 4;
        const int s  = edges[e];
        const int d  = edges[nEdges + e];
        const float4 v = *(const float4*)(X + (long long)s * 64 + c4);
        float* mp = msg + (long long)d * 64 + c4;
        unsafeAtomicAdd(mp + 0, v.x);
        unsafeAtomicAdd(mp + 1, v.y);
        unsafeAtomicAdd(mp + 2, v.z);
        unsafeAtomicAdd(mp + 3, v.w);
        if (cnt != nullptr && (idx & 15) == 0) {
            unsafeAtomicAdd(cnt + d, 1.0f);
        }
    }
}

// ---------------------------------------------------------------------------
// Fused SAGE node update, one wave per (16-node tile, 16-out-col tile):
//   OUT = act( (msg/cnt) @ Wl^T  +  Xin @ Wr^T  + bias )
// Uses V_WMMA_F32_16X16X4_F32 chained over K=64 (16 steps per matmul).
//
// A-fragment (16x4 f32, ISA 7.12.2):  lane L: M = L&15, VGPR j: K = 2*(L>>4)+j
// B-fragment (4x16 f32, mirrored):    lane L: N = L&15, VGPR j: K = 2*(L>>4)+j
// C/D (16x16 f32): VGPR i: M = i + 8*(L>>4), N = L&15
// ---------------------------------------------------------------------------
__global__ __launch_bounds__(256) void sage_gemm(
    const float* __restrict__ Xin,   // [N_NODES, 64] dense (root) input
    const float* __restrict__ msg,   // [N_NODES, 64] summed messages
    const float* __restrict__ cnt,   // [N_NODES] edge counts
    const float* __restrict__ Wl,    // [OUTD, 64] row-major
    const float* __restrict__ bias,  // [OUTD]
    const float* __restrict__ Wr,    // [OUTD, 64] row-major
    float* __restrict__ OUT,         // [N_NODES, OUTD]
    int OUTD, int applyRelu)
{
    const int nNTiles = OUTD >> 4;                 // 4 (layer1) or 2 (layer2)
    const int nMTiles = N_NODES >> 4;              // 3125
    const int waveInBlock = threadIdx.x >> 5;      // 8 waves / block
    const int wave = blockIdx.x * (blockDim.x >> 5) + waveInBlock;
    if (wave >= nMTiles * nNTiles) return;         // wave-uniform: EXEC stays ~0

    const int tm = wave / nNTiles;
    const int tn = wave - tm * nNTiles;

    const int lane = threadIdx.x & 31;
    const int r    = lane & 15;                    // A row / B col within tile
    const int kOff = (lane >> 4) << 1;             // 0 or 2

    const int m = tm * 16 + r;                     // node row for A loads
    const int n = tn * 16 + r;                     // out col for B loads

    const float inv = 1.0f / fmaxf(cnt[m], 1.0f);  // mean aggregation
    const float bv  = bias[tn * 16 + r];           // bias for this lane's N

    v8f c;
    #pragma unroll
    for (int i = 0; i < 8; ++i) c[i] = bv;         // seed accumulator with bias

    const float* msgRow = msg + (long long)m * 64;
    const float* xRow   = Xin + (long long)m * 64;
    const float* wlRow  = Wl  + (long long)n * 64;
    const float* wrRow  = Wr  + (long long)n * 64;

    // term 1: aggr @ Wl^T
    #pragma unroll
    for (int kg = 0; kg < 16; ++kg) {
        const int k = kg * 4 + kOff;
        v2f a, b;
        a.x = msgRow[k]     * inv;
        a.y = msgRow[k + 1] * inv;
        b.x = wlRow[k];
        b.y = wlRow[k + 1];
        c = __builtin_amdgcn_wmma_f32_16x16x4_f32(
                false, a, false, b, (short)0, c, false, false);
    }
    // term 2: Xin @ Wr^T
    #pragma unroll
    for (int kg = 0; kg < 16; ++kg) {
        const int k = kg * 4 + kOff;
        v2f a, b;
        a.x = xRow[k];
        a.y = xRow[k + 1];
        b.x = wrRow[k];
        b.y = wrRow[k + 1];
        c = __builtin_amdgcn_wmma_f32_16x16x4_f32(
                false, a, false, b, (short)0, c, false, false);
    }

    // Write back: VGPR i holds M = i + 8*(lane>>4), N = lane&15
    const int ncol  = lane & 15;
    const int mbase = tm * 16 + ((lane >> 4) << 3);
    #pragma unroll
    for (int i = 0; i < 8; ++i) {
        float v = c[i];
        if (applyRelu) v = fmaxf(v, 0.0f);
        OUT[(long long)(mbase + i) * OUTD + tn * 16 + ncol] = v;
    }
}

extern "C" void kernel_launch(void* const* d_in, const int* in_sizes, int n_in,
                              void* d_out, int out_size, void* d_ws, size_t ws_size,
                              hipStream_t stream) {
    const float* x   = (const float*)d_in[0];
    const int*   ei  = (const int*)  d_in[1];   // [2, N_EDGES]
    const float* Wl1 = (const float*)d_in[2];
    const float* bl1 = (const float*)d_in[3];
    const float* Wr1 = (const float*)d_in[4];
    const float* Wl2 = (const float*)d_in[5];
    const float* bl2 = (const float*)d_in[6];
    const float* Wr2 = (const float*)d_in[7];
    float* out = (float*)d_out;

    // Workspace layout (all fp32): msg[50000*64] | cnt[50000] | h[50000*64]
    float* msg = (float*)d_ws;
    float* cnt = msg + (size_t)N_NODES * 64;
    float* h   = cnt + N_NODES;

    const size_t msgBytes = (size_t)N_NODES * 64 * sizeof(float);
    const size_t cntBytes = (size_t)N_NODES * sizeof(float);

    // ---- Layer 1 ----
    hipMemsetAsync(msg, 0, msgBytes, stream);
    hipMemsetAsync(cnt, 0, cntBytes, stream);

    sage_scatter<<<4096, 256, 0, stream>>>(x, ei, msg, cnt, N_EDGES);

    {   // 3125 m-tiles * 4 n-tiles = 12500 waves, 8 waves/block
        const int waves  = (N_NODES / 16) * (HID_DIM / 16);
        const int blocks = (waves + 7) / 8;
        sage_gemm<<<blocks, 256, 0, stream>>>(x, msg, cnt, Wl1, bl1, Wr1,
                                              h, HID_DIM, /*relu=*/1);
    }

    // ---- Layer 2 ---- (same graph -> same counts; only re-zero msg)
    hipMemsetAsync(msg, 0, msgBytes, stream);

    sage_scatter<<<4096, 256, 0, stream>>>(h, ei, msg, nullptr, N_EDGES);

    {   // 3125 m-tiles * 2 n-tiles = 6250 waves
        const int waves  = (N_NODES / 16) * (OUT_DIM / 16);
        const int blocks = (waves + 7) / 8;
        sage_gemm<<<blocks, 256, 0, stream>>>(h, msg, cnt, Wl2, bl2, Wr2,
                                              out, OUT_DIM, /*relu=*/0);
    }
}